// EighMSE_2_MSE_42880953483666
// MI455X (gfx1250) — compile-verified
//
#include <hip/hip_runtime.h>
#include <math.h>
#include <stdint.h>

typedef __attribute__((ext_vector_type(2))) float v2f;
typedef __attribute__((ext_vector_type(8))) float v8f;

#define TPB 256
#define ROWS_PER_LANE 4
#define TILE_ROWS (TPB * ROWS_PER_LANE)   /* 1024 rows / tile           */
#define TILE_F4   (TILE_ROWS * 3 / 4)     /* 768 float4 per input array */
#define F4_PER_LANE (TILE_F4 / TPB)       /* 3 B128 async issues / lane */
#define MAXGRID 1024

// Closed-form 2x2 symmetric eigen: trace, disc = sqrt((a-c)^2+4b^2),
// and the rotation (cos t, sin t) that is the eigenvector of the LARGER
// eigenvalue (cos t >= 0 convention); small-eigvec = (-sin t, cos t).
__device__ __forceinline__ void sym2x2(float a, float b, float c,
                                       float& tr, float& disc,
                                       float& cs, float& sn) {
  tr = a + c;
  float h  = a - c;
  float b2 = b + b;
  disc = sqrtf(fmaf(h, h, b2 * b2));
  bool ok  = disc > 0.0f;
  float r  = ok ? 1.0f / disc : 0.0f;
  float c2 = ok ? h * r : 1.0f;           // cos(2t)
  float s2 = b2 * r;                      // sin(2t)
  float cc = fminf(fmaxf(0.5f * (1.0f + c2), 0.0f), 1.0f);
  float ss = fminf(fmaxf(0.5f * (1.0f - c2), 0.0f), 1.0f);
  cs = sqrtf(cc);
  sn = copysignf(sqrtf(ss), s2);
}

__device__ __forceinline__ void accum_row(float pa, float pb, float pc,
                                          float ta, float tb, float tc,
                                          float& aev, float& am,
                                          float& ac, float& as_) {
  float trp, dp, cp, sp_, trt, dt, ct, st_;
  sym2x2(pa, pb, pc, trp, dp, cp, sp_);
  sym2x2(ta, tb, tc, trt, dt, ct, st_);
  float du = trp - trt, dv = dp - dt;       // eval sq-err per row = (du^2+dv^2)/2
  aev = fmaf(du, du, aev);
  aev = fmaf(dv, dv, aev);
  float da = pa - ta, db = pb - tb, dc = pc - tc;
  am = fmaf(da, da, am);
  am = fmaf(db, db, am);
  am = fmaf(dc, dc, am);
  float dcs = cp - ct, dsn = sp_ - st_;
  ac  = fmaf(dcs, dcs, ac);
  as_ = fmaf(dsn, dsn, as_);
}

__global__ __launch_bounds__(TPB)
void eigh_loss_partials(const float* __restrict__ yp,
                        const float* __restrict__ yt,
                        float* __restrict__ part,
                        long long nrows, int nfull) {
  __shared__ float4 sP[2][TILE_F4];      // double-buffered pred tiles (24 KB)
  __shared__ float4 sT[2][TILE_F4];      // double-buffered true tiles (24 KB)
  __shared__ float wred[TPB / 32][4];

  const int t = threadIdx.x;
  float aev = 0.0f, am = 0.0f, ac = 0.0f, as_ = 0.0f;

  // LDS aperture: low 32 bits of the generic address are the LDS byte offset.
  const uint32_t ldsP[2] = {(uint32_t)(uintptr_t)(&sP[0][0]),
                            (uint32_t)(uintptr_t)(&sP[1][0])};
  const uint32_t ldsT[2] = {(uint32_t)(uintptr_t)(&sT[0][0]),
                            (uint32_t)(uintptr_t)(&sT[1][0])};

  // Issue one tile's worth of fully-coalesced async DMA (6 x B128 per lane,
  // tracked by ASYNCcnt) into LDS buffer `b`.
  auto issue_tile = [&](int tile, int b) {
    const float* gp = yp + (long long)tile * TILE_ROWS * 3;
    const float* gt = yt + (long long)tile * TILE_ROWS * 3;
#pragma unroll
    for (int k = 0; k < F4_PER_LANE; ++k) {
      uint32_t off16 = (uint32_t)(t + k * TPB) * 16u;
      asm volatile("global_load_async_to_lds_b128 %0, %1, %2"
                   :: "v"(ldsP[b] + off16), "v"(off16), "s"(gp) : "memory");
      asm volatile("global_load_async_to_lds_b128 %0, %1, %2"
                   :: "v"(ldsT[b] + off16), "v"(off16), "s"(gt) : "memory");
    }
  };

  // ---- double-buffered pipeline over full tiles ----
  int tile = blockIdx.x;
  bool have = tile < nfull;
  int cur = 0;
  if (have) issue_tile(tile, cur);

  while (have) {
    int ntile = tile + (int)gridDim.x;
    bool havenext = ntile < nfull;
    if (havenext) {
      issue_tile(ntile, cur ^ 1);                  // overlap DMA with compute
      int pf = ntile + (int)gridDim.x;
      if (pf < nfull) {                            // warm L2 two tiles ahead
        const float* nx = yp + (long long)pf * TILE_ROWS * 3;
        __builtin_prefetch(nx + (size_t)t * 12, 0, 1);
      }
      // 6 loads of the NEXT tile stay in flight; async completes in order,
      // so <=6 outstanding means the CURRENT buffer is fully resident.
      asm volatile("s_wait_asynccnt 6" ::: "memory");
    } else {
      asm volatile("s_wait_asynccnt 0" ::: "memory");
    }
    __syncthreads();                               // all waves' DMA visible

    // Lane handles 4 rows = 3 float4s from each staged tile.
    float4 p0 = sP[cur][3 * t + 0], p1 = sP[cur][3 * t + 1], p2 = sP[cur][3 * t + 2];
    float4 q0 = sT[cur][3 * t + 0], q1 = sT[cur][3 * t + 1], q2 = sT[cur][3 * t + 2];
    __syncthreads();                               // buffer reusable after this

    accum_row(p0.x, p0.y, p0.z, q0.x, q0.y, q0.z, aev, am, ac, as_);
    accum_row(p0.w, p1.x, p1.y, q0.w, q1.x, q1.y, aev, am, ac, as_);
    accum_row(p1.z, p1.w, p2.x, q1.z, q1.w, q2.x, aev, am, ac, as_);
    accum_row(p2.y, p2.z, p2.w, q2.y, q2.z, q2.w, aev, am, ac, as_);

    tile = ntile;
    have = havenext;
    cur ^= 1;
  }

  // Remainder rows (< TILE_ROWS): block 0 mops up with guarded direct loads.
  long long done = (long long)nfull * TILE_ROWS;
  if (blockIdx.x == 0 && done < nrows) {
    for (long long g = done + t; g < nrows; g += TPB) {
      const float* pp = yp + g * 3;
      const float* tt = yt + g * 3;
      accum_row(pp[0], pp[1], pp[2], tt[0], tt[1], tt[2], aev, am, ac, as_);
    }
  }

  // ---- wave32 reduction, first fold via v_wmma_f32_16x16x4_f32 ----
  // A = ones(16x4), B = (s, 0): D[m][n] = colsum(B)[n] = s[n] + s[n+16],
  // so D vgpr0 holds s[lane&15] + s[(lane&15)+16] in every lane.
  float vals[4] = {aev, am, ac, as_};
#if __has_builtin(__builtin_amdgcn_wmma_f32_16x16x4_f32)
  v2f ones; ones.x = 1.0f; ones.y = 1.0f;
  v8f cz = {0.f, 0.f, 0.f, 0.f, 0.f, 0.f, 0.f, 0.f};
#pragma unroll
  for (int i = 0; i < 4; ++i) {
    v2f bm; bm.x = vals[i]; bm.y = 0.0f;
    v8f d = __builtin_amdgcn_wmma_f32_16x16x4_f32(
        false, ones, false, bm, (short)0, cz, false, false);
    float v = d[0];
    v += __shfl_xor(v, 8, 32);
    v += __shfl_xor(v, 4, 32);
    v += __shfl_xor(v, 2, 32);
    v += __shfl_xor(v, 1, 32);
    vals[i] = v;
  }
#else
#pragma unroll
  for (int i = 0; i < 4; ++i) {
    float v = vals[i];
    v += __shfl_xor(v, 16, 32);
    v += __shfl_xor(v, 8, 32);
    v += __shfl_xor(v, 4, 32);
    v += __shfl_xor(v, 2, 32);
    v += __shfl_xor(v, 1, 32);
    vals[i] = v;
  }
#endif

  const int wave = t >> 5, lane = t & 31;
  if (lane == 0) {
    wred[wave][0] = vals[0]; wred[wave][1] = vals[1];
    wred[wave][2] = vals[2]; wred[wave][3] = vals[3];
  }
  __syncthreads();
  if (t < 4) {
    float s = 0.0f;
#pragma unroll
    for (int w2 = 0; w2 < TPB / 32; ++w2) s += wred[w2][t];
    part[(size_t)blockIdx.x * 4 + t] = s;   // deterministic per-block partials
  }
}

__global__ __launch_bounds__(TPB)
void eigh_loss_finalize(const float* __restrict__ part, int nparts,
                        const float* __restrict__ wts,
                        float* __restrict__ out, float invn) {
  __shared__ float red[4][TPB];
  const int t = threadIdx.x;
  float a0 = 0, a1 = 0, a2 = 0, a3 = 0;
  for (int i = t; i < nparts; i += TPB) {
    a0 += part[4 * i + 0]; a1 += part[4 * i + 1];
    a2 += part[4 * i + 2]; a3 += part[4 * i + 3];
  }
  red[0][t] = a0; red[1][t] = a1; red[2][t] = a2; red[3][t] = a3;
  __syncthreads();
  for (int s = TPB / 2; s > 0; s >>= 1) {
    if (t < s) {
      red[0][t] += red[0][t + s]; red[1][t] += red[1][t + s];
      red[2][t] += red[2][t + s]; red[3][t] += red[3][t + s];
    }
    __syncthreads();
  }
  if (t == 0) {
    float Aev = red[0][0], Am = red[1][0], Ac = red[2][0], As = red[3][0];
    float evals_mse = 0.25f * invn * Aev;            // sum(du^2+dv^2) / (4n)
    float mse_loss  = (invn / 3.0f) * Am;            // / (3n)
    float evec = (wts[1] + wts[4]) * (Ac * invn) +   // Mc channels
                 (wts[2] + wts[3]) * (As * invn);    // Ms channels
    out[0] = wts[0] * evals_mse + evec + wts[5] * mse_loss;
  }
}

extern "C" void kernel_launch(void* const* d_in, const int* in_sizes, int n_in,
                              void* d_out, int out_size, void* d_ws, size_t ws_size,
                              hipStream_t stream) {
  const float* yp = (const float*)d_in[0];
  const float* yt = (const float*)d_in[1];
  const float* w  = (const float*)d_in[2];
  long long n = (long long)in_sizes[0] / 3;
  int nfull = (int)(n / TILE_ROWS);                 // full tiles only
  int grid = nfull < 1 ? 1 : (nfull < MAXGRID ? nfull : MAXGRID);
  float* part = (float*)d_ws;                       // grid*4 floats (<= 16 KB)

  eigh_loss_partials<<<grid, TPB, 0, stream>>>(yp, yt, part, n, nfull);
  eigh_loss_finalize<<<1, TPB, 0, stream>>>(part, grid, w, (float*)d_out,
                                            n > 0 ? 1.0f / (float)n : 0.0f);
}